// DGCNN_4234837753922
// MI455X (gfx1250) — compile-verified
//
#include <hip/hip_runtime.h>
#include <hip/hip_bf16.h>

typedef __attribute__((ext_vector_type(2))) float v2f;
typedef __attribute__((ext_vector_type(8))) float v8f;

#define NPTS 2048
#define NGRAPH 16
#define KNN 20
#define NTOTAL (NPTS * NGRAPH)
#define SLOPE 0.02f

__device__ __forceinline__ v8f wmma_f32_4(v2f a, v2f b, v8f c) {
    // D = A(16x4 f32) * B(4x16 f32) + C(16x16 f32)
    return __builtin_amdgcn_wmma_f32_16x16x4_f32(
        /*neg_a=*/false, a, /*neg_b=*/false, b,
        /*c_mod=*/(short)0, c, /*reuse_a=*/false, /*reuse_b=*/false);
}

// ---------------------------------------------------------------------------
// kNN: per graph, for each row maintain the 20 smallest of
//      d'(i,j) = |x_j|^2 - 2 x_i . x_j   (|x_i|^2 constant per row -> dropped)
// Row block R=128 (one row per thread), col chunks of 64, distances via WMMA.
// ---------------------------------------------------------------------------
template <int C>
__global__ __launch_bounds__(128) void knn_kernel(const float* __restrict__ x,
                                                  int inC,
                                                  int* __restrict__ knn_out) {
    constexpr int R = 128, CCH = 64;
    extern __shared__ float smem[];
    float* sXr = smem;                 // R * C
    float* sXc = sXr + R * C;          // CCH * C
    float* sD  = sXc + CCH * C;        // R * CCH
    float* sX2 = sD + R * CCH;         // CCH

    const int tid  = threadIdx.x;
    const int lane = tid & 31;
    const int wid  = tid >> 5;
    const int l16  = lane & 15;
    const int lh   = lane >> 4;

    const int b       = blockIdx.x / (NPTS / R);
    const int rb      = blockIdx.x % (NPTS / R);
    const int rowBase = rb * R;
    const long gRow0  = (long)b * NPTS + rowBase;

    // stage the row panel (zero-pad features inC..C)
    for (int i = tid; i < R * C; i += 128) {
        int r = i / C, c = i % C;
        sXr[i] = (c < inC) ? x[(gRow0 + r) * inC + c] : 0.0f;
    }

    float bd[KNN];
    int   bi[KNN];
#pragma unroll
    for (int s = 0; s < KNN; ++s) { bd[s] = 3.0e38f; bi[s] = 0; }

    for (int cb = 0; cb < NPTS; cb += CCH) {
        __syncthreads();  // sD/sXc free for reuse; sXr ready on first pass
        for (int i = tid; i < CCH * C; i += 128) {
            int r = i / C, c = i % C;
            sXc[i] = (c < inC) ? x[((long)b * NPTS + cb + r) * inC + c] : 0.0f;
        }
        __syncthreads();
        if (tid < CCH) {
            float s = 0.0f;
            for (int c = 0; c < C; ++c) { float v = sXc[tid * C + c]; s += v * v; }
            sX2[tid] = s;
        }
        __syncthreads();

        // 32 tiles (8 M-tiles x 4 N-tiles), 8 per wave; EXEC all-ones here.
        for (int t = 0; t < 8; ++t) {
            const int tile = wid * 8 + t;
            const int tm = tile >> 2, tn = tile & 3;
            v8f acc;
#pragma unroll
            for (int q = 0; q < 8; ++q) acc[q] = 0.0f;
            const float* ar = &sXr[(tm * 16 + l16) * C + 2 * lh];
            const float* br = &sXc[(tn * 16 + l16) * C + 2 * lh];
            for (int kk = 0; kk < C; kk += 4) {
                v2f a;  a[0] = ar[kk];  a[1] = ar[kk + 1];
                v2f bb; bb[0] = br[kk]; bb[1] = br[kk + 1];
                acc = wmma_f32_4(a, bb, acc);
            }
            const float x2c = sX2[tn * 16 + l16];
#pragma unroll
            for (int q = 0; q < 8; ++q) {
                int rr = tm * 16 + q + 8 * lh;
                int cc = tn * 16 + l16;
                sD[rr * CCH + cc] = x2c - 2.0f * acc[q];
            }
        }
        __syncthreads();

        // top-K insertion (register arrays, fully static indices)
        for (int j = 0; j < CCH; ++j) {
            float dv = sD[tid * CCH + j];
            if (dv < bd[KNN - 1]) {
                bd[KNN - 1] = dv;
                bi[KNN - 1] = b * NPTS + cb + j;  // global neighbor index
#pragma unroll
                for (int s = KNN - 1; s > 0; --s) {
                    if (bd[s] < bd[s - 1]) {
                        float tf = bd[s]; bd[s] = bd[s - 1]; bd[s - 1] = tf;
                        int   ti = bi[s]; bi[s] = bi[s - 1]; bi[s - 1] = ti;
                    }
                }
            }
        }
    }

    const long orow = gRow0 + tid;
#pragma unroll
    for (int s = 0; s < KNN; ++s) knn_out[orow * KNN + s] = bi[s];
}

// ---------------------------------------------------------------------------
// EdgeConv: out[i] = max_k leakyrelu([x_i, x_j(k)-x_i] @ W + b)
// 16 points / block; K-dim (=KD, padded 2*CIN) GEMM via WMMA f32 16x16x4;
// N-tiles statically partitioned over NW waves; running max in registers.
// ---------------------------------------------------------------------------
template <int CIN, int KACT, int KD, int COUT, int NW>
__global__ __launch_bounds__(NW * 32) void edgeconv_kernel(
    const float* __restrict__ x, const int* __restrict__ knn,
    const float* __restrict__ W, const float* __restrict__ bias,
    float* __restrict__ out) {
    constexpr int P = 16;
    constexpr int NT = COUT / 16;
    constexpr int NTPW = NT / NW;
    static_assert(NT % NW == 0, "tile partition");

    __shared__ float sXi[P * CIN];
    __shared__ float sF[P * KD];

    const int tid  = threadIdx.x;
    const int lane = tid & 31;
    const int wid  = tid >> 5;
    const int l16  = lane & 15;
    const int lh   = lane >> 4;
    const long pBase = (long)blockIdx.x * P;

    for (int i = tid; i < P * CIN; i += NW * 32) {
        int r = i / CIN, c = i % CIN;
        sXi[i] = x[(pBase + r) * CIN + c];
    }
    __syncthreads();
    // constant halves of the feature tile: x_i and zero padding
    for (int i = tid; i < P * KD; i += NW * 32) {
        int r = i / KD, c = i % KD;
        if (c < CIN) sF[i] = sXi[r * CIN + c];
        else if (c >= 2 * CIN) sF[i] = 0.0f;
    }

    float hmax[NTPW * 8];
#pragma unroll
    for (int q = 0; q < NTPW * 8; ++q) hmax[q] = -3.0e38f;

    for (int k = 0; k < KNN; ++k) {
        __syncthreads();  // previous GEMM done reading sF
        for (int i = tid; i < P * CIN; i += NW * 32) {
            int r = i / CIN, c = i % CIN;
            int j = knn[(pBase + r) * KNN + k];
            sF[r * KD + CIN + c] = x[(long)j * CIN + c] - sXi[i];
        }
        __syncthreads();

        v8f acc[NTPW];
#pragma unroll
        for (int nt = 0; nt < NTPW; ++nt)
#pragma unroll
            for (int q = 0; q < 8; ++q) acc[nt][q] = 0.0f;

        const float* arow = &sF[l16 * KD + 2 * lh];
        for (int kk = 0; kk < KD; kk += 4) {
            v2f a; a[0] = arow[kk]; a[1] = arow[kk + 1];
            const int r0 = kk + 2 * lh;
#pragma unroll
            for (int nt = 0; nt < NTPW; ++nt) {
                const int col = (wid + nt * NW) * 16 + l16;
                v2f bb;
                bb[0] = (r0 < KACT)     ? W[r0 * COUT + col]       : 0.0f;
                bb[1] = (r0 + 1 < KACT) ? W[(r0 + 1) * COUT + col] : 0.0f;
                acc[nt] = wmma_f32_4(a, bb, acc[nt]);
            }
        }
#pragma unroll
        for (int nt = 0; nt < NTPW; ++nt) {
            const int col = (wid + nt * NW) * 16 + l16;
            const float bv = bias[col];
#pragma unroll
            for (int q = 0; q < 8; ++q) {
                float h = acc[nt][q] + bv;
                h = (h > 0.0f) ? h : (SLOPE * h);
                float m = hmax[nt * 8 + q];
                hmax[nt * 8 + q] = (h > m) ? h : m;
            }
        }
    }

#pragma unroll
    for (int nt = 0; nt < NTPW; ++nt) {
        const int col = (wid + nt * NW) * 16 + l16;
#pragma unroll
        for (int q = 0; q < 8; ++q) {
            const long row = pBase + q + 8 * lh;
            out[row * COUT + col] = hmax[nt * 8 + q];
        }
    }
}

// ---------------------------------------------------------------------------
// Global max pool over concat([x1,x2,x3,x4]) -> g[B,512]
// ---------------------------------------------------------------------------
__global__ void pool_kernel(const float* __restrict__ x1, const float* __restrict__ x2,
                            const float* __restrict__ x3, const float* __restrict__ x4,
                            float* __restrict__ g) {
    const int b = blockIdx.x;
    const int f = threadIdx.x;  // 512 threads
    const float* src;
    int stride;
    if (f < 64)        { src = x1 + (long)b * NPTS * 64  + f;         stride = 64; }
    else if (f < 128)  { src = x2 + (long)b * NPTS * 64  + (f - 64);  stride = 64; }
    else if (f < 256)  { src = x3 + (long)b * NPTS * 128 + (f - 128); stride = 128; }
    else               { src = x4 + (long)b * NPTS * 256 + (f - 256); stride = 256; }
    float m = -3.0e38f;
    for (int p = 0; p < NPTS; ++p) m = fmaxf(m, src[(long)p * stride]);
    g[b * 512 + f] = m;
}

// ---------------------------------------------------------------------------
// Final dense: relu(g @ Wl + bl) -> [16,1024]
// ---------------------------------------------------------------------------
__global__ void dense_kernel(const float* __restrict__ g, const float* __restrict__ Wl,
                             const float* __restrict__ bl, float* __restrict__ out) {
    const int o = blockIdx.x * blockDim.x + threadIdx.x;  // 16*1024
    const int b = o >> 10, c = o & 1023;
    float s = bl[c];
    for (int q = 0; q < 512; ++q) s += g[b * 512 + q] * Wl[q * 1024 + c];
    out[o] = fmaxf(s, 0.0f);
}

extern "C" void kernel_launch(void* const* d_in, const int* in_sizes, int n_in,
                              void* d_out, int out_size, void* d_ws, size_t ws_size,
                              hipStream_t stream) {
    (void)in_sizes; (void)n_in; (void)out_size; (void)ws_size;
    const float* pos = (const float*)d_in[0];
    // d_in[1] = batch (int64) unused: equal-sized sorted graphs == reshape
    const float* W1 = (const float*)d_in[2];  const float* b1 = (const float*)d_in[3];
    const float* W2 = (const float*)d_in[4];  const float* b2 = (const float*)d_in[5];
    const float* W3 = (const float*)d_in[6];  const float* b3 = (const float*)d_in[7];
    const float* W4 = (const float*)d_in[8];  const float* b4 = (const float*)d_in[9];
    const float* Wl = (const float*)d_in[10]; const float* bl = (const float*)d_in[11];
    float* outp = (float*)d_out;

    char* ws = (char*)d_ws;
    float* x1  = (float*)(ws);                                  // 32768*64*4  =  8 MB
    float* x2  = (float*)(ws + 8388608ull);                     //  8 MB
    float* x3  = (float*)(ws + 16777216ull);                    // 32768*128*4 = 16 MB
    float* x4  = (float*)(ws + 33554432ull);                    // 32768*256*4 = 32 MB
    int*   idx = (int*)  (ws + 67108864ull);                    // 32768*20*4  ~ 2.6 MB
    float* g   = (float*)(ws + 69730304ull);                    // 16*512*4

    constexpr int R = 128, CCH = 64;
    const size_t smem4   = (size_t)(R * 4   + CCH * 4   + R * CCH + CCH) * 4;
    const size_t smem64  = (size_t)(R * 64  + CCH * 64  + R * CCH + CCH) * 4;
    const size_t smem128 = (size_t)(R * 128 + CCH * 128 + R * CCH + CCH) * 4;
    const dim3 knnGrid(NGRAPH * (NPTS / R));
    const dim3 ecGrid(NTOTAL / 16);

    // Layer 1: knn on pos (C=3 padded to 4), EdgeConv 6->64 (K padded to 8)
    knn_kernel<4><<<knnGrid, 128, smem4, stream>>>(pos, 3, idx);
    edgeconv_kernel<3, 6, 8, 64, 4><<<ecGrid, 128, 0, stream>>>(pos, idx, W1, b1, x1);

    // Layer 2: knn on x1 (64), EdgeConv 128->64
    knn_kernel<64><<<knnGrid, 128, smem64, stream>>>(x1, 64, idx);
    edgeconv_kernel<64, 128, 128, 64, 4><<<ecGrid, 128, 0, stream>>>(x1, idx, W2, b2, x2);

    // Layer 3: knn on x2 (64), EdgeConv 128->128
    knn_kernel<64><<<knnGrid, 128, smem64, stream>>>(x2, 64, idx);
    edgeconv_kernel<64, 128, 128, 128, 8><<<ecGrid, 256, 0, stream>>>(x2, idx, W3, b3, x3);

    // Layer 4: knn on x3 (128), EdgeConv 256->256
    knn_kernel<128><<<knnGrid, 128, smem128, stream>>>(x3, 128, idx);
    edgeconv_kernel<128, 256, 256, 256, 8><<<ecGrid, 256, 0, stream>>>(x3, idx, W4, b4, x4);

    // Pool + final MLP
    pool_kernel<<<NGRAPH, 512, 0, stream>>>(x1, x2, x3, x4, g);
    dense_kernel<<<64, 256, 0, stream>>>(g, Wl, bl, outp);
}